// SpatialTemporalAttention_62036507623584
// MI455X (gfx1250) — compile-verified
//
#include <hip/hip_runtime.h>

// Problem constants (from reference): B=8, C=256, T=16, H=W=32
#define B_DIM   8
#define C_DIM   256
#define T_DIM   16
#define HW_DIM  1024
#define CR_DIM  16
#define BC_TOT  (B_DIM * C_DIM)       // 2048
#define PER_BC  (T_DIM * HW_DIM)      // 16384

typedef float v2f __attribute__((ext_vector_type(2)));
typedef float v4f __attribute__((ext_vector_type(4)));
typedef float v8f __attribute__((ext_vector_type(8)));

// ---------------------------------------------------------------------------
// Kernel 1: one pass over x producing
//   sp[bc][hw] = mean over T     (spatial pooled)
//   tp[bc][t]  = mean over HW    (temporal pooled)
//   cp[bc]     = mean over T,HW  (channel pooled)
// One block per (b,c): 16384 contiguous floats. x loads are regular-temporal
// on purpose: 128 MB of x fits in the 192 MB L2 and is re-read by the apply
// pass.
// ---------------------------------------------------------------------------
__global__ __launch_bounds__(256) void pool_kernel(const float* __restrict__ x,
                                                   float* __restrict__ sp,
                                                   float* __restrict__ tp,
                                                   float* __restrict__ cp) {
  const int  bc   = blockIdx.x;
  const long base = (long)bc * PER_BC;
  const int  tid  = threadIdx.x;
  const int  lane = tid & 31;
  const int  wave = tid >> 5;

  __shared__ float red[8][T_DIM];   // per-wave partial HW-sums per t

  float a0 = 0.f, a1 = 0.f, a2 = 0.f, a3 = 0.f;   // per-thread T-sums (4 hw each)
  for (int t = 0; t < T_DIM; ++t) {
    const v4f v = *(const v4f*)(x + base + (long)t * HW_DIM + tid * 4);
    a0 += v.x; a1 += v.y; a2 += v.z; a3 += v.w;
    float s = v.x + v.y + v.z + v.w;
    #pragma unroll
    for (int off = 16; off > 0; off >>= 1) s += __shfl_xor(s, off, 32);
    if (lane == 0) red[wave][t] = s;
  }
  __syncthreads();

  v4f o;
  o.x = a0 * (1.f / 16.f); o.y = a1 * (1.f / 16.f);
  o.z = a2 * (1.f / 16.f); o.w = a3 * (1.f / 16.f);
  *(v4f*)(sp + (long)bc * HW_DIM + tid * 4) = o;

  if (tid < T_DIM) {
    float s = 0.f;
    #pragma unroll
    for (int w = 0; w < 8; ++w) s += red[w][tid];
    tp[bc * T_DIM + tid] = s * (1.f / 1024.f);
    float c = s;                    // reduce the 16 per-t sums across lanes 0..15
    #pragma unroll
    for (int off = 8; off > 0; off >>= 1) c += __shfl_xor(c, off, 32);
    if (tid == 0) cp[bc] = c * (1.f / 16384.f);
  }
}

// ---------------------------------------------------------------------------
// Kernel 2: SE branch on WMMA (f32, 16x16x4), one wave per 16-position tile.
//   H = relu(W1[16,256] @ X[256,16] + b1)      (64 WMMAs, K=256)
//   A = sigmoid(W2[256,16] @ H[16,16] + b2)    (16 row-tiles x 4 WMMAs)
// Attention written IN PLACE over X (columns fully consumed before overwrite).
// X element addressing: Xb[c*cs + p*ps], Xb = X + b*baseStride.
// ---------------------------------------------------------------------------
__global__ __launch_bounds__(32) void se_wmma_kernel(
    float* X,
    const float* __restrict__ W1, const float* __restrict__ B1,
    const float* __restrict__ W2, const float* __restrict__ B2,
    int cs, int ps, long baseStride, int tilesPerB, int validPerB) {
  const int tile = blockIdx.x;
  const int b    = tile / tilesPerB;
  const int p0   = (tile % tilesPerB) * 16;
  const int lane = threadIdx.x;      // 0..31, wave32
  const int hs   = lane >> 4;        // lane half: selects K pair (A/B) / M half (C,D)
  const int ln   = lane & 15;        // M (A-frag) or N (B/C/D-frag)

  float* Xb = X + (long)b * baseStride;

  const int  p  = p0 + ln;
  const bool pv = (p < validPerB);
  const int  pc = pv ? p : 0;        // clamp address for inactive (padded) columns

  __shared__ float h1[CR_DIM][17];   // hidden tile [k][n], padded stride

  // ---- GEMM1: 16x256 @ 256x16 -> 16x16, K in steps of 4 ----
  v8f acc = {};
  for (int kb = 0; kb < C_DIM; kb += 4) {
    const int ka = kb + hs * 2;
    v2f av, bv;
    // A (16x4 f32): lanes 0-15 -> K=0,1 ; lanes 16-31 -> K=2,3 ; row M = ln
    av.x = W1[ln * C_DIM + ka];
    av.y = W1[ln * C_DIM + ka + 1];
    // B (4x16 f32): row K striped across lanes, N = ln
    bv.x = Xb[(long)ka * cs + (long)pc * ps];
    bv.y = Xb[(long)(ka + 1) * cs + (long)pc * ps];
    if (!pv) { bv.x = 0.f; bv.y = 0.f; }
    acc = __builtin_amdgcn_wmma_f32_16x16x4_f32(
        false, av, false, bv, (short)0, acc, false, false);
  }
  // bias + relu; C layout: lane ln holds column n=ln, rows m = hs*8 + j
  #pragma unroll
  for (int j = 0; j < 8; ++j) {
    const int m = hs * 8 + j;
    float v = acc[j] + B1[m];
    h1[m][ln] = v > 0.f ? v : 0.f;
  }
  __syncthreads();   // single-wave WG: lowers to wave barrier, orders LDS via DScnt

  // ---- GEMM2: 256x16 @ 16x16 -> 256x16, 16 row tiles ----
  for (int mt = 0; mt < 16; ++mt) {
    v8f acc2 = {};
    #pragma unroll
    for (int kb = 0; kb < CR_DIM; kb += 4) {
      const int ka = kb + hs * 2;
      v2f av, bv;
      av.x = W2[(mt * 16 + ln) * CR_DIM + ka];
      av.y = W2[(mt * 16 + ln) * CR_DIM + ka + 1];
      bv.x = h1[ka][ln];
      bv.y = h1[ka + 1][ln];
      acc2 = __builtin_amdgcn_wmma_f32_16x16x4_f32(
          false, av, false, bv, (short)0, acc2, false, false);
    }
    #pragma unroll
    for (int j = 0; j < 8; ++j) {
      const int row = mt * 16 + hs * 8 + j;
      float v = acc2[j] + B2[row];
      v = 1.f / (1.f + __expf(-v));           // sigmoid
      if (pv) Xb[(long)row * cs + (long)pc * ps] = v;
    }
  }
}

// ---------------------------------------------------------------------------
// Kernel 3: out = x * att_s[hw] * att_t[t] * att_c  (one block per (b,c)).
// out is streamed with NON-TEMPORAL stores so the 128 MB output does not
// evict x (L2-resident from the pool pass) from the 192 MB L2 — x re-reads
// stay on-chip, HBM traffic drops from ~384 MB to ~256 MB.
// ---------------------------------------------------------------------------
__global__ __launch_bounds__(256) void apply_kernel(const float* __restrict__ x,
                                                    const float* __restrict__ att_s,
                                                    const float* __restrict__ att_t,
                                                    const float* __restrict__ att_c,
                                                    float* __restrict__ out) {
  const int  bc   = blockIdx.x;
  const long base = (long)bc * PER_BC;
  const int  tid  = threadIdx.x;

  __shared__ float at[T_DIM];
  __shared__ __align__(16) float as[HW_DIM];

  const float ac = att_c[bc];
  if (tid < T_DIM) at[tid] = att_t[bc * T_DIM + tid] * ac;
  *(v4f*)(as + tid * 4) = *(const v4f*)(att_s + (long)bc * HW_DIM + tid * 4);
  __syncthreads();

  const v4f sv = *(const v4f*)(as + tid * 4);
  for (int t = 0; t < T_DIM; ++t) {
    const float m  = at[t];
    const v4f   xv = *(const v4f*)(x + base + (long)t * HW_DIM + tid * 4);
    v4f ov = xv * sv * m;
    __builtin_nontemporal_store(ov, (v4f*)(out + base + (long)t * HW_DIM + tid * 4));
  }
}

// ---------------------------------------------------------------------------
extern "C" void kernel_launch(void* const* d_in, const int* in_sizes, int n_in,
                              void* d_out, int out_size, void* d_ws, size_t ws_size,
                              hipStream_t stream) {
  const float* x   = (const float*)d_in[0];
  const float* ws1 = (const float*)d_in[1];
  const float* bs1 = (const float*)d_in[2];
  const float* ws2 = (const float*)d_in[3];
  const float* bs2 = (const float*)d_in[4];
  const float* wt1 = (const float*)d_in[5];
  const float* bt1 = (const float*)d_in[6];
  const float* wt2 = (const float*)d_in[7];
  const float* bt2 = (const float*)d_in[8];
  const float* wc1 = (const float*)d_in[9];
  const float* bc1 = (const float*)d_in[10];
  const float* wc2 = (const float*)d_in[11];
  const float* bc2 = (const float*)d_in[12];
  float* out = (float*)d_out;

  // Workspace layout (floats): sp[2048*1024] | tp[2048*16] | cp[2048]
  float* wsf = (float*)d_ws;
  float* sp  = wsf;
  float* tp  = wsf + (size_t)BC_TOT * HW_DIM;
  float* cp  = tp  + (size_t)BC_TOT * T_DIM;

  // 1) Pool all three tensors in one pass over x.
  pool_kernel<<<BC_TOT, 256, 0, stream>>>(x, sp, tp, cp);

  // 2) SE branches on WMMA; attention overwrites the pooled buffers in place.
  //    spatial: N = 8*1024 positions, X[c][p] = sp[b*262144 + c*1024 + p]
  se_wmma_kernel<<<B_DIM * (HW_DIM / 16), 32, 0, stream>>>(
      sp, ws1, bs1, ws2, bs2,
      /*cs=*/HW_DIM, /*ps=*/1, /*baseStride=*/(long)C_DIM * HW_DIM,
      /*tilesPerB=*/HW_DIM / 16, /*validPerB=*/HW_DIM);
  //    temporal: N = 8*16, X[c][t] = tp[b*4096 + c*16 + t]
  se_wmma_kernel<<<B_DIM, 32, 0, stream>>>(
      tp, wt1, bt1, wt2, bt2,
      /*cs=*/T_DIM, /*ps=*/1, /*baseStride=*/(long)C_DIM * T_DIM,
      /*tilesPerB=*/1, /*validPerB=*/T_DIM);
  //    channel: N = 8 (padded to 16), X[c][b] = cp[b*256 + c]
  se_wmma_kernel<<<1, 32, 0, stream>>>(
      cp, wc1, bc1, wc2, bc2,
      /*cs=*/1, /*ps=*/C_DIM, /*baseStride=*/0,
      /*tilesPerB=*/1, /*validPerB=*/B_DIM);

  // 3) Fused elementwise apply (NT stores for out).
  apply_kernel<<<BC_TOT, 256, 0, stream>>>(x, sp, tp, cp, out);
}